// SourceBasedGT_TGN_SocialViral_13769665151525
// MI455X (gfx1250) — compile-verified
//
#include <hip/hip_runtime.h>
#include <math.h>

typedef float v2f __attribute__((ext_vector_type(2)));
typedef float v8f __attribute__((ext_vector_type(8)));

__device__ __forceinline__ v8f wmma_f32(v2f a, v2f b, v8f c) {
  // D = A(16x4) * B(4x16) + C(16x16), all f32, wave32
  return __builtin_amdgcn_wmma_f32_16x16x4_f32(false, a, false, b, (short)0, c,
                                               false, false);
}

// ---------------------------------------------------------------------------
// Generic 16x16-tile GEMM:  C[row, col] = act( A @ W^T + bias (+ res) )
// A: [M x K] row-major (lda), W: [Ncols x K] row-major (ldw)
// grid.x = M/16 row tiles, grid.y = N/16 col tiles, block = 32 (one wave)
// act: 0 = none, 1 = exact GELU
// ---------------------------------------------------------------------------
__global__ void gemm32(const float* __restrict__ A, int lda,
                       const float* __restrict__ W, int ldw,
                       const float* __restrict__ bias,
                       const float* __restrict__ res,
                       float* __restrict__ C, int ldc, int K, int act) {
  const int lane = threadIdx.x & 31;
  const int half = lane >> 4;
  const int l15  = lane & 15;
  const int row0 = blockIdx.x * 16;
  const int col0 = blockIdx.y * 16;
  const int r = row0 + l15;   // A row this lane feeds
  const int n = col0 + l15;   // output column this lane feeds
  v8f acc = {};
  for (int k0 = 0; k0 < K; k0 += 4) {
    const int kk = k0 + (half << 1);
    v2f a, b;
    a.x = A[r * lda + kk];
    a.y = A[r * lda + kk + 1];
    b.x = W[n * ldw + kk];        // B(k, n) = W[n][k]
    b.y = W[n * ldw + kk + 1];
    acc = wmma_f32(a, b, acc);
  }
  const float bv = bias ? bias[n] : 0.0f;
#pragma unroll
  for (int j = 0; j < 8; ++j) {
    const int row = row0 + j + half * 8;
    float v = acc[j] + bv;
    if (res) v += res[row * ldc + n];
    if (act == 1) v = 0.5f * v * (1.0f + erff(v * 0.70710678118654752f));
    C[row * ldc + n] = v;
  }
}

// ---------------------------------------------------------------------------
// LayerNorm over width 32 (one row per wave-of-32 lanes), in place.
// ---------------------------------------------------------------------------
__global__ void ln32(float* __restrict__ X, const float* __restrict__ g,
                     const float* __restrict__ be, int nrows) {
  const int lane = threadIdx.x & 31;
  const int row  = (blockIdx.x * blockDim.x + threadIdx.x) >> 5;
  if (row >= nrows) return;
  float x = X[row * 32 + lane];
  float s = x;
  for (int m = 16; m >= 1; m >>= 1) s += __shfl_xor(s, m, 32);
  const float mu = s * (1.0f / 32.0f);
  const float d  = x - mu;
  float v = d * d;
  for (int m = 16; m >= 1; m >>= 1) v += __shfl_xor(v, m, 32);
  X[row * 32 + lane] = d * rsqrtf(v * (1.0f / 32.0f) + 1e-5f) * g[lane] + be[lane];
}

// ---------------------------------------------------------------------------
// Flash attention, one wave per (16-row query tile, head). HD = 8, N = 2048.
// qkv: [N x 96] = [q(32) | k(32) | v(32)], ao: [N x 32]
// ---------------------------------------------------------------------------
__global__ void attn(const float* __restrict__ qkv, float* __restrict__ ao) {
  const int lane = threadIdx.x & 31;
  const int half = lane >> 4;
  const int l15  = lane & 15;
  const int row0 = blockIdx.x * 16;
  const int h    = blockIdx.y;
  const int qoff = h * 8, koff = 32 + h * 8, voff = 64 + h * 8;
  const int kk   = half << 1;
  __shared__ float P[16 * 16];

  // Q A-tiles for k-steps 0 and 4 (K-dim = HD = 8)
  v2f qa0, qa1;
  {
    const float* qr = qkv + (row0 + l15) * 96 + qoff;
    qa0.x = qr[kk];     qa0.y = qr[kk + 1];
    qa1.x = qr[4 + kk]; qa1.y = qr[4 + kk + 1];
  }
  float rm[8], rl[8];
#pragma unroll
  for (int j = 0; j < 8; ++j) { rm[j] = -1e30f; rl[j] = 0.0f; }
  v8f acc = {};

  for (int jt = 0; jt < 128; ++jt) {
    if (jt + 1 < 128) {  // prefetch next K/V tiles
      __builtin_prefetch(qkv + ((jt + 1) * 16 + l15) * 96 + koff, 0, 1);
      __builtin_prefetch(qkv + ((jt + 1) * 16 + l15) * 96 + voff, 0, 1);
    }
    // S = (Q K^T) * 1/sqrt(8)
    v8f s = {};
    {
      const float* kr = qkv + (jt * 16 + l15) * 96 + koff;  // B(k=d, n=key)
      v2f b0, b1;
      b0.x = kr[kk];     b0.y = kr[kk + 1];
      b1.x = kr[4 + kk]; b1.y = kr[4 + kk + 1];
      s = wmma_f32(qa0, b0, s);
      s = wmma_f32(qa1, b1, s);
    }
    // online softmax; row (j + 8*half) lives across the 16 lanes of a half
#pragma unroll
    for (int j = 0; j < 8; ++j) {
      float sv = s[j] * 0.35355339059327373f;
      float mj = sv;
      for (int m = 8; m >= 1; m >>= 1) mj = fmaxf(mj, __shfl_xor(mj, m, 16));
      const float mnew = fmaxf(rm[j], mj);
      const float corr = expf(rm[j] - mnew);
      const float p    = expf(sv - mnew);
      float ps = p;
      for (int m = 8; m >= 1; m >>= 1) ps += __shfl_xor(ps, m, 16);
      rl[j] = rl[j] * corr + ps;
      rm[j] = mnew;
      acc[j] *= corr;
      P[(j + half * 8) * 16 + l15] = p;   // C layout -> LDS row-major
    }
    __syncthreads();
    // acc += P(16x16) @ V(16x8, zero-padded to 16)  -> 4 WMMA k-steps
#pragma unroll
    for (int ks = 0; ks < 16; ks += 4) {
      v2f a, b;
      a.x = P[l15 * 16 + ks + kk];       // A-layout reload from LDS
      a.y = P[l15 * 16 + ks + kk + 1];
      const float* vr = qkv + (jt * 16 + ks + kk) * 96 + voff;
      if (l15 < 8) { b.x = vr[l15]; b.y = vr[96 + l15]; }
      else         { b.x = 0.0f;    b.y = 0.0f; }
      acc = wmma_f32(a, b, acc);
    }
    __syncthreads();
  }
  if (l15 < 8) {
#pragma unroll
    for (int j = 0; j < 8; ++j) {
      const int row = row0 + j + half * 8;
      ao[row * 32 + qoff + l15] = acc[j] / rl[j];
    }
  }
}

// ---------------------------------------------------------------------------
// SCN: s_out = relu( LN16( x2 @ W_scn^T + b_scn ) ), fused (row fits a half)
// ---------------------------------------------------------------------------
__global__ void scn(const float* __restrict__ X, const float* __restrict__ W,
                    const float* __restrict__ bias, const float* __restrict__ g,
                    const float* __restrict__ be, float* __restrict__ S) {
  const int lane = threadIdx.x & 31;
  const int half = lane >> 4;
  const int l15  = lane & 15;
  const int row0 = blockIdx.x * 16;
  v8f acc = {};
  for (int k0 = 0; k0 < 32; k0 += 4) {
    const int kk = k0 + (half << 1);
    v2f a, b;
    a.x = X[(row0 + l15) * 32 + kk]; a.y = X[(row0 + l15) * 32 + kk + 1];
    b.x = W[l15 * 32 + kk];          b.y = W[l15 * 32 + kk + 1];
    acc = wmma_f32(a, b, acc);
  }
  const float gv = g[l15], bev = be[l15], bv = bias[l15];
#pragma unroll
  for (int j = 0; j < 8; ++j) {
    const float x = acc[j] + bv;
    float s = x;
    for (int m = 8; m >= 1; m >>= 1) s += __shfl_xor(s, m, 16);
    const float mu = s * (1.0f / 16.0f);
    const float d  = x - mu;
    float v = d * d;
    for (int m = 8; m >= 1; m >>= 1) v += __shfl_xor(v, m, 16);
    float y = d * rsqrtf(v * (1.0f / 16.0f) + 1e-5f) * gv + bev;
    S[(row0 + j + half * 8) * 16 + l15] = fmaxf(y, 0.0f);
  }
}

__global__ void copy_f32(const float* __restrict__ src, float* __restrict__ dst,
                         int n) {
  int i = blockIdx.x * blockDim.x + threadIdx.x;
  if (i < n) dst[i] = src[i];
}

// ---------------------------------------------------------------------------
// Sequential TGN memory scan (exact jax.lax.scan semantics incl. collisions)
// One block of 32 lanes; lane = memory dimension. LDS row-cache by node id.
// ---------------------------------------------------------------------------
__global__ void tgn_scan(const int* __restrict__ src, const int* __restrict__ dst,
                         const float* __restrict__ ef,
                         const float* __restrict__ mem_in,
                         const float* __restrict__ Wih, const float* __restrict__ bih,
                         const float* __restrict__ Whh, const float* __restrict__ bhh,
                         const float* __restrict__ Wm1, const float* __restrict__ bm1,
                         const float* __restrict__ Wm2, const float* __restrict__ bm2,
                         float* __restrict__ mem_out) {
  const int lane = threadIdx.x;  // 0..31
  __shared__ float msgs[16][32];
  __shared__ float buf[80];
  __shared__ float h1[32];
  __shared__ int   cidx[40];
  __shared__ float cval[40][32];
  __shared__ int   ncache_s;
  if (lane == 0) ncache_s = 0;
  __syncthreads();

  // messages from PRE-update memory
  for (int b = 0; b < 16; ++b) {
    const int s = src[b], d = dst[b];
    buf[lane]      = mem_in[s * 32 + lane];
    buf[32 + lane] = mem_in[d * 32 + lane];
    if (lane < 3) buf[64 + lane] = ef[b * 3 + lane];
    __syncthreads();
    float a = bm1[lane];
    for (int k = 0; k < 67; ++k) a += Wm1[lane * 67 + k] * buf[k];
    h1[lane] = fmaxf(a, 0.0f);
    __syncthreads();
    float a2 = bm2[lane];
    for (int k = 0; k < 32; ++k) a2 += Wm2[lane * 32 + k] * h1[k];
    msgs[b][lane] = a2;
    __syncthreads();
  }

  // sequential scan: for each edge, GRU-update src row then dst row
  for (int b = 0; b < 16; ++b) {
    const int idxs[2] = { src[b], dst[b] };
    for (int t = 0; t < 2; ++t) {
      const int r = idxs[t];
      const int nc = ncache_s;
      int e = -1;
      for (int q = 0; q < nc; ++q) if (cidx[q] == r) e = q;
      const float hk = (e >= 0) ? cval[e][lane] : mem_in[r * 32 + lane];
      buf[lane] = hk;
      __syncthreads();
      float ir = bih[lane], iz = bih[32 + lane], inn = bih[64 + lane];
      for (int k = 0; k < 32; ++k) {
        const float mk = msgs[b][k];
        ir  += Wih[lane * 32 + k] * mk;
        iz  += Wih[(32 + lane) * 32 + k] * mk;
        inn += Wih[(64 + lane) * 32 + k] * mk;
      }
      float hr = bhh[lane], hz = bhh[32 + lane], hn = bhh[64 + lane];
      for (int k = 0; k < 32; ++k) {
        const float hv = buf[k];
        hr += Whh[lane * 32 + k] * hv;
        hz += Whh[(32 + lane) * 32 + k] * hv;
        hn += Whh[(64 + lane) * 32 + k] * hv;
      }
      const float rg = 1.0f / (1.0f + expf(-(ir + hr)));
      const float zg = 1.0f / (1.0f + expf(-(iz + hz)));
      const float ng = tanhf(inn + rg * hn);
      const float hnew = (1.0f - zg) * ng + zg * hk;
      __syncthreads();
      if (e < 0) {
        e = nc;
        if (lane == 0) { cidx[e] = r; ncache_s = nc + 1; }
      }
      __syncthreads();
      cval[e][lane] = hnew;
      __syncthreads();
    }
  }
  const int nc = ncache_s;
  for (int q = 0; q < nc; ++q) mem_out[cidx[q] * 32 + lane] = cval[q][lane];
}

// ---------------------------------------------------------------------------
// Tiny edge-scoring MLP head: one lane per edge (B = 16)
// ---------------------------------------------------------------------------
__global__ void edge_head(const float* __restrict__ S, const int* __restrict__ src,
                          const int* __restrict__ dst,
                          const float* __restrict__ W1, const float* __restrict__ b1,
                          const float* __restrict__ W2, const float* __restrict__ b2,
                          const float* __restrict__ W3, const float* __restrict__ b3,
                          float* __restrict__ out) {
  const int e = threadIdx.x;
  if (e >= 16) return;
  float h0[32], h1v[32], h2v[16];
  const int s = src[e], d = dst[e];
  for (int i = 0; i < 16; ++i) { h0[i] = S[s * 16 + i]; h0[16 + i] = S[d * 16 + i]; }
  for (int o = 0; o < 32; ++o) {
    float a = b1[o];
    for (int k = 0; k < 32; ++k) a += W1[o * 32 + k] * h0[k];
    h1v[o] = fmaxf(a, 0.0f);
  }
  for (int o = 0; o < 16; ++o) {
    float a = b2[o];
    for (int k = 0; k < 32; ++k) a += W2[o * 32 + k] * h1v[k];
    h2v[o] = fmaxf(a, 0.0f);
  }
  float a = b3[0];
  for (int k = 0; k < 16; ++k) a += W3[k] * h2v[k];
  out[e] = a;
}

extern "C" void kernel_launch(void* const* d_in, const int* in_sizes, int n_in,
                              void* d_out, int out_size, void* d_ws, size_t ws_size,
                              hipStream_t stream) {
  const int*   src   = (const int*)  d_in[0];
  const int*   dst   = (const int*)  d_in[1];
  const float* ef    = (const float*)d_in[2];
  const float* nf    = (const float*)d_in[4];   // node_features [2048 x 128]
  const float* mem   = (const float*)d_in[6];   // memory [2048 x 32]
  const float* Wemb  = (const float*)d_in[7];
  const float* bemb  = (const float*)d_in[8];
  const float* Wqkv  = (const float*)d_in[9];
  const float* bqkv  = (const float*)d_in[10];
  const float* Wo    = (const float*)d_in[11];
  const float* bo    = (const float*)d_in[12];
  const float* g1    = (const float*)d_in[13];
  const float* be1   = (const float*)d_in[14];
  const float* g2    = (const float*)d_in[15];
  const float* be2   = (const float*)d_in[16];
  const float* Wf1   = (const float*)d_in[17];
  const float* bf1   = (const float*)d_in[18];
  const float* Wf2   = (const float*)d_in[19];
  const float* bf2   = (const float*)d_in[20];
  const float* Wscn  = (const float*)d_in[21];
  const float* bscn  = (const float*)d_in[22];
  const float* gscn  = (const float*)d_in[23];
  const float* bescn = (const float*)d_in[24];
  const float* We1   = (const float*)d_in[25];
  const float* bE1   = (const float*)d_in[26];
  const float* We2   = (const float*)d_in[27];
  const float* bE2   = (const float*)d_in[28];
  const float* We3   = (const float*)d_in[29];
  const float* bE3   = (const float*)d_in[30];
  const float* Wih   = (const float*)d_in[31];
  const float* bih   = (const float*)d_in[32];
  const float* Whh   = (const float*)d_in[33];
  const float* bhh   = (const float*)d_in[34];
  const float* Wm1   = (const float*)d_in[35];
  const float* bm1   = (const float*)d_in[36];
  const float* Wm2   = (const float*)d_in[37];
  const float* bm2   = (const float*)d_in[38];

  float* out_logit = (float*)d_out;        // [16]
  float* out_mem   = (float*)d_out + 16;   // [2048 x 32]

  float* ws  = (float*)d_ws;
  float* x0  = ws;              // 65536   node embeddings
  float* qkv = ws + 65536;      // 196608  (region later reused by fh/x2)
  float* fh  = ws + 65536;      // 131072  FFN hidden (qkv dead by then)
  float* x2  = ws + 196608;     // 65536
  float* ao  = ws + 262144;     // 65536   attention output
  float* x1  = ws + 327680;     // 65536
  float* so  = ws + 393216;     // 32768   s_out
  // total: 425984 floats = 1.70 MB

  // ---- memory path: new_mem = scan(memory) ----
  copy_f32<<<(65536 + 255) / 256, 256, 0, stream>>>(mem, out_mem, 65536);
  tgn_scan<<<1, 32, 0, stream>>>(src, dst, ef, mem, Wih, bih, Whh, bhh,
                                 Wm1, bm1, Wm2, bm2, out_mem);

  // ---- embedding + transformer (batch-invariant, computed once) ----
  gemm32<<<dim3(128, 2), 32, 0, stream>>>(nf, 128, Wemb, 128, bemb, nullptr,
                                          x0, 32, 128, 0);
  gemm32<<<dim3(128, 6), 32, 0, stream>>>(x0, 32, Wqkv, 32, bqkv, nullptr,
                                          qkv, 96, 32, 0);
  attn<<<dim3(128, 4), 32, 0, stream>>>(qkv, ao);
  gemm32<<<dim3(128, 2), 32, 0, stream>>>(ao, 32, Wo, 32, bo, x0,
                                          x1, 32, 32, 0);
  ln32<<<256, 256, 0, stream>>>(x1, g1, be1, 2048);
  gemm32<<<dim3(128, 4), 32, 0, stream>>>(x1, 32, Wf1, 32, bf1, nullptr,
                                          fh, 64, 32, 1);
  gemm32<<<dim3(128, 2), 32, 0, stream>>>(fh, 64, Wf2, 64, bf2, x1,
                                          x2, 32, 64, 0);
  ln32<<<256, 256, 0, stream>>>(x2, g2, be2, 2048);
  scn<<<128, 32, 0, stream>>>(x2, Wscn, bscn, gscn, bescn, so);
  edge_head<<<1, 32, 0, stream>>>(so, src, dst, We1, bE1, We2, bE2, We3, bE3,
                                  out_logit);
}